// CSRA_head_14018773254582
// MI455X (gfx1250) — compile-verified
//
#include <hip/hip_runtime.h>

typedef __attribute__((ext_vector_type(2))) float v2f;
typedef __attribute__((ext_vector_type(8))) float v8f;
typedef __attribute__((ext_vector_type(4))) int   v4i;

#define B_ 64
#define S_ 576
#define D_ 1024
#define C_ 100

#define ROWS 32            // s-rows per block
#define LDS_STRIDE 1028    // padded row stride (floats): conflict-free A-fragment reads

#if defined(__has_builtin)
#  if __has_builtin(__builtin_amdgcn_global_load_async_to_lds_b128)
#    define HAVE_ASYNC_LDS 1
#  endif
#endif
#ifndef HAVE_ASYNC_LDS
#  define HAVE_ASYNC_LDS 0
#endif

// ---------------------------------------------------------------------------
// Kernel 0: zero the [B,C] accumulator in workspace (harness poisons ws).
// ---------------------------------------------------------------------------
__global__ void csra_zero(float* __restrict__ p, int n) {
    int i = blockIdx.x * 256 + threadIdx.x;
    if (i < n) p[i] = 0.0f;
}

// ---------------------------------------------------------------------------
// Kernel 1: main fused kernel.
//   logits = patch @ conv_w^T        (fp32 WMMA 16x16x4)
//   acc[b,c] += sum_s sigmoid(logits + conv_b) * rowsum[b,s]
// One block = (b, 32-row s-tile). 8 waves: 0-6 = 16-class N-tiles, 7 = rowsums.
// A-tile staged to LDS via CDNA5 async global->LDS (ASYNCcnt) when available.
// ---------------------------------------------------------------------------
__global__ __launch_bounds__(256)
void csra_main(const float* __restrict__ patch,
               const float* __restrict__ conv_w,
               const float* __restrict__ conv_b,
               float* __restrict__ acc) {
    __shared__ float ldsA[ROWS * LDS_STRIDE];
    __shared__ float ldsRS[ROWS];

    const int blk  = blockIdx.x;                 // 0 .. B*(S/ROWS)-1
    const int b    = blk / (S_ / ROWS);
    const int s0   = (blk % (S_ / ROWS)) * ROWS;
    const float* gA = patch + ((size_t)b * S_ + s0) * D_;

    const int t = threadIdx.x;

    // Stage 32 rows x 1024 floats -> LDS (padded rows), coalesced b128 per lane.
#if HAVE_ASYNC_LDS
    typedef __attribute__((address_space(1))) v4i* g4p;
    typedef __attribute__((address_space(3))) v4i* l4p;
    #pragma unroll
    for (int i = 0; i < ROWS; ++i) {
        __builtin_amdgcn_global_load_async_to_lds_b128(
            (g4p)(size_t)(gA + (size_t)i * D_ + 4 * t),
            (l4p)(size_t)(&ldsA[i * LDS_STRIDE + 4 * t]),
            0, 0);
    }
#  if __has_builtin(__builtin_amdgcn_s_wait_asynccnt)
    __builtin_amdgcn_s_wait_asynccnt(0);
#  else
    asm volatile("s_wait_asynccnt 0x0" ::: "memory");
#  endif
#else
    #pragma unroll
    for (int i = 0; i < ROWS; ++i) {
        const float4 v = *(const float4*)(gA + (size_t)i * D_ + 4 * t);
        *(float4*)(&ldsA[i * LDS_STRIDE + 4 * t]) = v;
    }
#endif
    __syncthreads();

    const int wave = t >> 5;
    const int lane = t & 31;
    const int lm   = lane & 15;     // M (for A) / N (for B) within fragment
    const int hf   = lane >> 4;     // lane-half selects K offset (+2)

    const int n0 = wave * 16;
    const int c  = n0 + lm;
    const int cw = (c < C_) ? c : (C_ - 1);      // clamp OOB classes (masked later)

    v8f acc0 = {0,0,0,0,0,0,0,0};                // rows s0+0..15  x classes n0..n0+15
    v8f acc1 = {0,0,0,0,0,0,0,0};                // rows s0+16..31 x classes n0..n0+15

    if (wave == 7) {
        // Rowsums: lane -> row, float4 LDS reads (16B aligned: 1028*4 % 16 == 0)
        float s = 0.0f;
        const float* rp = &ldsA[lane * LDS_STRIDE];
        for (int k = 0; k < D_; k += 4) {
            float4 v = *(const float4*)(rp + k);
            s += v.x + v.y + v.z + v.w;
        }
        ldsRS[lane] = s;
    } else {
        // A-fragment (16x4 f32): lane holds patch[m=lm][k0 + 2*hf + {0,1}]
        // B-fragment (4x16 f32): lane holds conv_w[c=n0+lm][k0 + 2*hf + {0,1}]
        const float* lA0 = &ldsA[lm * LDS_STRIDE + 2 * hf];
        const float* lA1 = &ldsA[(lm + 16) * LDS_STRIDE + 2 * hf];
        const float* pB  = conv_w + (size_t)cw * D_ + 2 * hf;

        for (int k = 0; k < D_; k += 32) {
            #pragma unroll
            for (int kk = 0; kk < 32; kk += 4) {
                v2f av0 = *(const v2f*)(lA0 + k + kk);
                v2f av1 = *(const v2f*)(lA1 + k + kk);
                v2f bv  = *(const v2f*)(pB  + k + kk);
                acc0 = __builtin_amdgcn_wmma_f32_16x16x4_f32(
                    false, av0, false, bv, (short)0, acc0, false, false);
                acc1 = __builtin_amdgcn_wmma_f32_16x16x4_f32(
                    false, av1, false, bv, (short)0, acc1, false, false);
            }
        }
    }
    __syncthreads();

    if (wave < 7) {
        const float bias = conv_b[cw];
        // C/D layout: VGPR j, lane-half hf -> row m = j + 8*hf, col n = lm
        float p = 0.0f;
        #pragma unroll
        for (int j = 0; j < 8; ++j) {
            const int m0 = j + 8 * hf;           // row in first 16-row tile
            float l0 = acc0[j] + bias;
            float l1 = acc1[j] + bias;
            float sg0 = 1.0f / (1.0f + __expf(-l0));
            float sg1 = 1.0f / (1.0f + __expf(-l1));
            p += sg0 * ldsRS[m0] + sg1 * ldsRS[m0 + 16];
        }
        // lanes l and l+16 hold the same class -> fold halves
        p += __shfl_xor(p, 16, 32);
        if (hf == 0 && c < C_) {
            atomicAdd(&acc[b * C_ + c], p);
        }
    }
}

// ---------------------------------------------------------------------------
// Kernel 2: finalize. out[b,c] = class_token[b]·fc_w[c] + fc_b[c]
//                                + lam * acc[b,c] / (S*D)
// One wave per (b,c); 8 waves per block; grid = B*C/8 = 800.
// ---------------------------------------------------------------------------
__global__ __launch_bounds__(256)
void csra_final(const float* __restrict__ cls,
                const float* __restrict__ fc_w,
                const float* __restrict__ fc_b,
                const float* __restrict__ lam,
                const float* __restrict__ acc,
                float* __restrict__ out) {
    const int wave = threadIdx.x >> 5;
    const int lane = threadIdx.x & 31;
    const int idx  = blockIdx.x * 8 + wave;      // 0 .. B*C-1
    if (idx >= B_ * C_) return;
    const int b = idx / C_;
    const int c = idx % C_;
    const float* x = cls  + (size_t)b * D_;
    const float* w = fc_w + (size_t)c * D_;
    float s = 0.0f;
    for (int k = lane; k < D_; k += 32) s += x[k] * w[k];
    #pragma unroll
    for (int off = 16; off > 0; off >>= 1) s += __shfl_xor(s, off, 32);
    if (lane == 0) {
        out[idx] = s + fc_b[c] + lam[0] * acc[idx] * (1.0f / ((float)S_ * (float)D_));
    }
}

// ---------------------------------------------------------------------------
extern "C" void kernel_launch(void* const* d_in, const int* in_sizes, int n_in,
                              void* d_out, int out_size, void* d_ws, size_t ws_size,
                              hipStream_t stream) {
    (void)in_sizes; (void)n_in; (void)out_size; (void)ws_size;
    const float* patch  = (const float*)d_in[0];
    const float* cls    = (const float*)d_in[1];
    const float* conv_w = (const float*)d_in[2];
    const float* conv_b = (const float*)d_in[3];
    const float* fc_w   = (const float*)d_in[4];
    const float* fc_b   = (const float*)d_in[5];
    const float* lam    = (const float*)d_in[6];
    float* accum = (float*)d_ws;                 // [B*C] fp32 accumulator
    float* out   = (float*)d_out;

    csra_zero<<<(B_ * C_ + 255) / 256, 256, 0, stream>>>(accum, B_ * C_);
    csra_main<<<B_ * (S_ / ROWS), 256, 0, stream>>>(patch, conv_w, conv_b, accum);
    csra_final<<<(B_ * C_) / 8, 256, 0, stream>>>(cls, fc_w, fc_b, lam, accum, out);
}